// QuantumBlock_67637144977641
// MI455X (gfx1250) — compile-verified
//
#include <hip/hip_runtime.h>

typedef __attribute__((ext_vector_type(2))) float v2f;
typedef __attribute__((ext_vector_type(8))) float v8f;

#define NQ       14
#define NSTATE   16384      // 2^14
#define NLAYERS  6
#define BATCH    1024
#define THREADS  256        // 8 wave32

__device__ __forceinline__ float2 cmul(float2 a, float2 b) {
  return make_float2(a.x * b.x - a.y * b.y, a.x * b.y + a.y * b.x);
}

// Build the 4 fused 16x16 complex group unitaries from the 2x2 gates in LDS.
// gates[slot*4 + r*2 + c], slot==14 is identity. Group g acts on bits
// [bbase+3 .. bbase], bit (bbase+p) corresponds to wire 13-(bbase+p).
__device__ __forceinline__ void build_gmat(const float2* gates, float2* gmat, int tid) {
  const int swire[4][4] = {{13, 12, 11, 10}, {9, 8, 7, 6}, {5, 4, 3, 2}, {14, 14, 1, 0}};
  const int r = tid >> 4, c = tid & 15;
#pragma unroll
  for (int g = 0; g < 4; ++g) {
    float2 acc = make_float2(1.f, 0.f);
#pragma unroll
    for (int p = 0; p < 4; ++p) {
      const int rp = (r >> p) & 1, cp = (c >> p) & 1;
      acc = cmul(acc, gates[swire[g][p] * 4 + rp * 2 + cp]);
    }
    gmat[g * 256 + r * 16 + c] = acc;
  }
}

// Apply 16x16 complex unitary G over bit group [bb+3..bb] of the state.
// src/dst are 16384-entry complex LDS buffers. Complex GEMM via 4 real
// 16x16x16 products, each as 4 chained V_WMMA_F32_16X16X4_F32.
__device__ __forceinline__ void apply_group(const float2* __restrict__ src,
                                            float2* __restrict__ dst,
                                            const float2* __restrict__ G,
                                            int bb, int tid) {
  const int lane = tid & 31;
  const int wave = tid >> 5;
  const int half = lane >> 4;   // 0: lanes 0-15, 1: lanes 16-31
  const int q    = lane & 15;

  // A = G in WMMA 16x4-chunk layout: reg j of chunk cc holds K=4*cc+j+2*half, M=q.
  v2f ar[4], ai[4], an[4];
#pragma unroll
  for (int cc = 0; cc < 4; ++cc) {
#pragma unroll
    for (int j = 0; j < 2; ++j) {
      const int k = 4 * cc + j + 2 * half;
      const float2 a = G[q * 16 + k];
      ar[cc][j] = a.x;
      ai[cc][j] = a.y;
      an[cc][j] = -a.y;
    }
  }

  const int lomask = (1 << bb) - 1;
#pragma unroll
  for (int t = 0; t < 8; ++t) {                 // 8 tiles per wave, 64 tiles total
    const int colbase = (wave * 8 + t) * 16;
    const int col = colbase + q;                // this lane's column (N = q)
    const int lo = col & lomask;
    const int hi = col >> bb;
    const int base = (hi << (bb + 4)) | lo;

    // B = state tile: reg j of chunk cc holds row K=4*cc+j+2*half, col N=q.
    v2f br[4], bi[4];
#pragma unroll
    for (int cc = 0; cc < 4; ++cc)
#pragma unroll
      for (int j = 0; j < 2; ++j) {
        const int k = 4 * cc + j + 2 * half;
        const float2 v = src[base + (k << bb)];
        br[cc][j] = v.x;
        bi[cc][j] = v.y;
      }

    v8f dr = {0.f, 0.f, 0.f, 0.f, 0.f, 0.f, 0.f, 0.f};
    v8f di = {0.f, 0.f, 0.f, 0.f, 0.f, 0.f, 0.f, 0.f};
#pragma unroll
    for (int cc = 0; cc < 4; ++cc) {
      // Dr += Ar*Br + (-Ai)*Bi ; Di += Ar*Bi + Ai*Br
      dr = __builtin_amdgcn_wmma_f32_16x16x4_f32(false, ar[cc], false, br[cc], (short)0, dr, false, false);
      dr = __builtin_amdgcn_wmma_f32_16x16x4_f32(false, an[cc], false, bi[cc], (short)0, dr, false, false);
      di = __builtin_amdgcn_wmma_f32_16x16x4_f32(false, ar[cc], false, bi[cc], (short)0, di, false, false);
      di = __builtin_amdgcn_wmma_f32_16x16x4_f32(false, ai[cc], false, br[cc], (short)0, di, false, false);
    }

    // D layout: VGPR v -> M = v + 8*half, N = q
#pragma unroll
    for (int v = 0; v < 8; ++v) {
      const int m = v + 8 * half;
      dst[base + (m << bb)] = make_float2(dr[v], di[v]);
    }
  }
}

__global__ void __launch_bounds__(THREADS)
qsim_kernel(const float* __restrict__ x, const float* __restrict__ wts,
            float* __restrict__ out) {
  extern __shared__ float2 smem[];
  float2* stA   = smem;                 // 16384
  float2* stB   = smem + NSTATE;        // 16384
  float2* gmat  = smem + 2 * NSTATE;    // 4 groups * 256
  float2* gates = gmat + 1024;          // 16 slots * 4 entries
  float*  red   = (float*)(gates + 64); // 256 floats

  const int tid  = threadIdx.x;
  const int bidx = blockIdx.x;
  const int bbase[4] = {0, 4, 8, 10};

  // |0...0>
  for (int i = tid; i < NSTATE; i += THREADS)
    stA[i] = make_float2(i == 0 ? 1.f : 0.f, 0.f);

  // encoding gates: RX(x_w) per wire, slot 14 = identity
  if (tid < NQ) {
    const float th = x[bidx * NQ + tid];
    float s, c;
    sincosf(th * 0.5f, &s, &c);
    gates[tid * 4 + 0] = make_float2(c, 0.f);
    gates[tid * 4 + 1] = make_float2(0.f, -s);
    gates[tid * 4 + 2] = make_float2(0.f, -s);
    gates[tid * 4 + 3] = make_float2(c, 0.f);
  } else if (tid == NQ) {
    gates[14 * 4 + 0] = make_float2(1.f, 0.f);
    gates[14 * 4 + 1] = make_float2(0.f, 0.f);
    gates[14 * 4 + 2] = make_float2(0.f, 0.f);
    gates[14 * 4 + 3] = make_float2(1.f, 0.f);
  }
  __syncthreads();
  build_gmat(gates, gmat, tid);
  __syncthreads();

  float2* cur = stA;
  float2* nxt = stB;
#pragma unroll
  for (int g = 0; g < 4; ++g) {
    apply_group(cur, nxt, gmat + g * 256, bbase[g], tid);
    __syncthreads();
    float2* tmp = cur; cur = nxt; nxt = tmp;
  }

  for (int l = 0; l < NLAYERS; ++l) {
    // Rot(phi,theta,omega) = RZ(omega) RY(theta) RZ(phi)
    if (tid < NQ) {
      const float* wp = wts + (l * NQ + tid) * 3;
      const float phi = wp[0], theta = wp[1], omega = wp[2];
      float s, c, sa, ca, sm, cm;
      sincosf(theta * 0.5f, &s, &c);
      sincosf((phi + omega) * 0.5f, &sa, &ca);
      sincosf((phi - omega) * 0.5f, &sm, &cm);
      gates[tid * 4 + 0] = make_float2(c * ca, -c * sa);   //  c e^{-i(phi+om)/2}
      gates[tid * 4 + 1] = make_float2(-s * cm, -s * sm);  // -s e^{+i(phi-om)/2}
      gates[tid * 4 + 2] = make_float2(s * cm, -s * sm);   //  s e^{-i(phi-om)/2}
      gates[tid * 4 + 3] = make_float2(c * ca, c * sa);    //  c e^{+i(phi+om)/2}
    }
    __syncthreads();
    build_gmat(gates, gmat, tid);
    __syncthreads();
#pragma unroll
    for (int g = 0; g < 4; ++g) {
      apply_group(cur, nxt, gmat + g * 256, bbase[g], tid);
      __syncthreads();
      float2* tmp = cur; cur = nxt; nxt = tmp;
    }

    // CNOT ring: sequential CNOT(w, (w+r)%NQ) is a pure index permutation
    const int r = (l % (NQ - 1)) + 1;
    for (int i = tid; i < NSTATE; i += THREADS) {
      int v = i;
#pragma unroll
      for (int w = 0; w < NQ; ++w) {
        const int cpos = 13 - w;
        const int tpos = 13 - ((w + r) % NQ);
        v ^= ((v >> cpos) & 1) << tpos;
      }
      nxt[v] = cur[i];
    }
    __syncthreads();
    float2* tmp = cur; cur = nxt; nxt = tmp;
  }

  // <Z_0>: wire 0 is bit 13 of the flat index
  float s = 0.f;
  for (int i = tid; i < NSTATE; i += THREADS) {
    const float2 a = cur[i];
    const float p = a.x * a.x + a.y * a.y;
    s += ((i >> 13) & 1) ? -p : p;
  }
  red[tid] = s;
  __syncthreads();
  for (int off = THREADS / 2; off > 0; off >>= 1) {
    if (tid < off) red[tid] += red[tid + off];
    __syncthreads();
  }
  if (tid == 0) out[bidx] = red[0];
}

extern "C" void kernel_launch(void* const* d_in, const int* in_sizes, int n_in,
                              void* d_out, int out_size, void* d_ws, size_t ws_size,
                              hipStream_t stream) {
  const float* x   = (const float*)d_in[0];   // (1024, 14) f32
  const float* wts = (const float*)d_in[1];   // (6, 14, 3) f32
  float* out = (float*)d_out;                 // (1024,) f32

  const size_t smem_bytes =
      (size_t)(2 * NSTATE + 1024 + 64) * sizeof(float2) + THREADS * sizeof(float);
  (void)hipFuncSetAttribute((const void*)qsim_kernel,
                            hipFuncAttributeMaxDynamicSharedMemorySize,
                            (int)smem_bytes);
  qsim_kernel<<<BATCH, THREADS, smem_bytes, stream>>>(x, wts, out);
}